// DySAT_44220983280298
// MI455X (gfx1250) — compile-verified
//
#include <hip/hip_runtime.h>
#include <hip/hip_bf16.h>

typedef float v2f __attribute__((ext_vector_type(2)));
typedef float v8f __attribute__((ext_vector_type(8)));

#define NEG_SLOPE 0.2f
#define GAT_EPS 1e-16f
// order-preserving key of -inf
#define ORD_NEG_INF 0x007FFFFFu

__device__ __forceinline__ unsigned f2ord(float f) {
    unsigned u = __float_as_uint(f);
    return (u & 0x80000000u) ? ~u : (u | 0x80000000u);
}
__device__ __forceinline__ float ord2f(unsigned u) {
    return __uint_as_float((u & 0x80000000u) ? (u & 0x7FFFFFFFu) : ~u);
}

// ---------------------------------------------------------------------------
// GEMM: Hout[n_rows,128] = X[n_rows,128] @ W[128,128]  (f32 WMMA 16x16x4)
// 256 threads = 8 wave32; block = 16-row M tile; wave w -> N tile [16w,16w+16)
// ---------------------------------------------------------------------------
__global__ void gat_gemm_wmma(const float* __restrict__ X,
                              const float* __restrict__ W,
                              float* __restrict__ Hout, int n_rows) {
    __shared__ float lA[16 * 33];    // padded stride 33 (bank-conflict free)
    __shared__ float lW[32 * 128];   // 16 KB K-chunk of W

    const int wave = threadIdx.x >> 5;
    const int lane = threadIdx.x & 31;
    const int m0 = blockIdx.x * 16;
    const int n0 = wave * 16;
    const int ml = lane & 15;
    const int khalf = (lane < 16) ? 0 : 2;

    v8f acc = {0.f, 0.f, 0.f, 0.f, 0.f, 0.f, 0.f, 0.f};

    for (int kc = 0; kc < 128; kc += 32) {
        // stage A tile 16x32
        for (int t = threadIdx.x; t < 16 * 32; t += 256) {
            int r = t >> 5, c = t & 31;
            int row = m0 + r;
            lA[r * 33 + c] = (row < n_rows) ? X[(size_t)row * 128 + kc + c] : 0.f;
        }
        // stage W chunk 32x128 (contiguous, coalesced)
        for (int t = threadIdx.x; t < 32 * 128; t += 256) {
            lW[t] = W[(size_t)kc * 128 + t];
        }
        __syncthreads();

        #pragma unroll
        for (int k = 0; k < 32; k += 4) {
            const int kk = k + khalf;
            v2f a, b;
            // A 16x4: lanes 0-15 hold M=ml {K=k,k+1}; lanes 16-31 {K=k+2,k+3}
            a.x = lA[ml * 33 + kk];
            a.y = lA[ml * 33 + kk + 1];
            // B 4x16: lanes 0-15 hold N=ml {K=k,k+1}; lanes 16-31 {K=k+2,k+3}
            b.x = lW[kk * 128 + n0 + ml];
            b.y = lW[(kk + 1) * 128 + n0 + ml];
            acc = __builtin_amdgcn_wmma_f32_16x16x4_f32(
                false, a, false, b, (short)0, acc, false, false);
        }
        __syncthreads();
    }

    // C/D layout: lanes 0-15 -> rows 0..7 col=lane; lanes 16-31 -> rows 8..15
    const int col = ml;
    const int rbase = (lane < 16) ? 0 : 8;
    #pragma unroll
    for (int r = 0; r < 8; ++r) {
        int row = m0 + rbase + r;
        if (row < n_rows) Hout[(size_t)row * 128 + n0 + col] = acc[r];
    }
}

// ---------------------------------------------------------------------------
// Per-node attention scalars: a_src[n,h] = <h[n,h,:], att_src[h,:]>, same dst
// ---------------------------------------------------------------------------
__global__ void gat_att(const float* __restrict__ Hbuf,
                        const float* __restrict__ att_s,
                        const float* __restrict__ att_d,
                        float* __restrict__ a_s, float* __restrict__ a_d, int n) {
    int i = blockIdx.x * blockDim.x + threadIdx.x;
    if (i >= n * 8) return;
    int node = i >> 3, hh = i & 7;
    const float* hp = Hbuf + (size_t)node * 128 + hh * 16;
    float s = 0.f, d = 0.f;
    #pragma unroll
    for (int c = 0; c < 16; ++c) {
        float v = hp[c];
        s += v * att_s[hh * 16 + c];
        d += v * att_d[hh * 16 + c];
    }
    a_s[i] = s;
    a_d[i] = d;
}

// init max-keys, sums, and the N*128 accumulator (every call; buffers poisoned)
__global__ void gat_init(unsigned* __restrict__ mb, float* __restrict__ ss,
                         float* __restrict__ acc, int n) {
    int i = blockIdx.x * blockDim.x + threadIdx.x;
    if (i < n * 8) { mb[i] = ORD_NEG_INF; ss[i] = 0.f; }
    if (i < n * 128) acc[i] = 0.f;
}

__device__ __forceinline__ float edge_logit(const float* a_s, const float* a_d,
                                            int s, int d, int hh) {
    float el = a_s[s * 8 + hh] + a_d[d * 8 + hh];
    return el > 0.f ? el : NEG_SLOPE * el;
}

// segment max over incoming edges (order-preserving uint atomicMax)
__global__ void gat_edge_max(const int* __restrict__ src, const int* __restrict__ dst,
                             const float* __restrict__ a_s, const float* __restrict__ a_d,
                             unsigned* __restrict__ mb, int e) {
    int i = blockIdx.x * blockDim.x + threadIdx.x;
    if (i >= e * 8) return;
    int eid = i >> 3, hh = i & 7;
    int s = src[eid], d = dst[eid];
    float el = edge_logit(a_s, a_d, s, d, hh);
    atomicMax(&mb[d * 8 + hh], f2ord(el));
}

// segment sum of exp(e - m)
__global__ void gat_edge_sum(const int* __restrict__ src, const int* __restrict__ dst,
                             const float* __restrict__ a_s, const float* __restrict__ a_d,
                             const unsigned* __restrict__ mb, float* __restrict__ ss, int e) {
    int i = blockIdx.x * blockDim.x + threadIdx.x;
    if (i >= e * 8) return;
    int eid = i >> 3, hh = i & 7;
    int s = src[eid], d = dst[eid];
    float el = edge_logit(a_s, a_d, s, d, hh);
    float ee = __expf(el - ord2f(mb[d * 8 + hh]));
    atomicAdd(&ss[d * 8 + hh], ee);
}

// alpha-weighted scatter of 16-float head message into L2-resident accumulator
__global__ void gat_edge_scatter(const int* __restrict__ src, const int* __restrict__ dst,
                                 const float* __restrict__ a_s, const float* __restrict__ a_d,
                                 const unsigned* __restrict__ mb, const float* __restrict__ ss,
                                 const float* __restrict__ Hbuf, float* __restrict__ acc, int e) {
    int i = blockIdx.x * blockDim.x + threadIdx.x;
    if (i >= e * 8) return;
    int eid = i >> 3, hh = i & 7;
    int s = src[eid], d = dst[eid];
    float el = edge_logit(a_s, a_d, s, d, hh);
    float ee = __expf(el - ord2f(mb[d * 8 + hh]));
    float alpha = ee / (ss[d * 8 + hh] + GAT_EPS);
    const float* hp = Hbuf + (size_t)s * 128 + hh * 16;
    float* op = acc + (size_t)d * 128 + hh * 16;
    #pragma unroll
    for (int c = 0; c < 16; ++c) atomicAdd(&op[c], hp[c] * alpha);
}

// out = (acc + bias) [optionally ReLU], in place or to another buffer
__global__ void gat_bias_act(const float* __restrict__ acc, const float* __restrict__ bias,
                             float* __restrict__ outp, int n, int do_relu) {
    int i = blockIdx.x * blockDim.x + threadIdx.x;
    if (i >= n * 128) return;
    float v = acc[i] + bias[i & 127];
    if (do_relu) v = v > 0.f ? v : 0.f;
    outp[i] = v;
}

// ---------------------------------------------------------------------------
extern "C" void kernel_launch(void* const* d_in, const int* in_sizes, int n_in,
                              void* d_out, int out_size, void* d_ws, size_t ws_size,
                              hipStream_t stream) {
    const float* x   = (const float*)d_in[0];
    const int*   ei  = (const int*)d_in[1];
    const float* W1  = (const float*)d_in[2];
    const float* as1 = (const float*)d_in[3];
    const float* ad1 = (const float*)d_in[4];
    const float* b1  = (const float*)d_in[5];
    const float* W2  = (const float*)d_in[6];
    const float* as2 = (const float*)d_in[7];
    const float* ad2 = (const float*)d_in[8];
    const float* b2  = (const float*)d_in[9];
    float* out = (float*)d_out;

    const int N = in_sizes[0] / 128;
    const int E = in_sizes[1] / 2;
    const int* src = ei;
    const int* dst = ei + E;

    // workspace carve-up
    float*    bufH  = (float*)d_ws;                      // N*128  (h of layer)
    float*    bufAc = bufH + (size_t)N * 128;            // N*128  (layer-1 accum)
    float*    a_s   = bufAc + (size_t)N * 128;           // N*8
    float*    a_d   = a_s + (size_t)N * 8;               // N*8
    unsigned* mb    = (unsigned*)(a_d + (size_t)N * 8);  // N*8
    float*    ss    = (float*)(mb + (size_t)N * 8);      // N*8

    const int gN128 = (N * 128 + 255) / 256;
    const int gNH   = (N * 8 + 255) / 256;
    const int gEH   = (E * 8 + 255) / 256;
    const int gGemm = (N + 15) / 16;

    // ---- layer 1 ----
    gat_gemm_wmma<<<gGemm, 256, 0, stream>>>(x, W1, bufH, N);
    gat_att<<<gNH, 256, 0, stream>>>(bufH, as1, ad1, a_s, a_d, N);
    gat_init<<<gN128, 256, 0, stream>>>(mb, ss, bufAc, N);
    gat_edge_max<<<gEH, 256, 0, stream>>>(src, dst, a_s, a_d, mb, E);
    gat_edge_sum<<<gEH, 256, 0, stream>>>(src, dst, a_s, a_d, mb, ss, E);
    gat_edge_scatter<<<gEH, 256, 0, stream>>>(src, dst, a_s, a_d, mb, ss, bufH, bufAc, E);
    gat_bias_act<<<gN128, 256, 0, stream>>>(bufAc, b1, bufAc, N, 1);  // ReLU in place

    // ---- layer 2 (aggregate straight into d_out) ----
    gat_gemm_wmma<<<gGemm, 256, 0, stream>>>(bufAc, W2, bufH, N);
    gat_att<<<gNH, 256, 0, stream>>>(bufH, as2, ad2, a_s, a_d, N);
    gat_init<<<gN128, 256, 0, stream>>>(mb, ss, out, N);
    gat_edge_max<<<gEH, 256, 0, stream>>>(src, dst, a_s, a_d, mb, E);
    gat_edge_sum<<<gEH, 256, 0, stream>>>(src, dst, a_s, a_d, mb, ss, E);
    gat_edge_scatter<<<gEH, 256, 0, stream>>>(src, dst, a_s, a_d, mb, ss, bufH, out, E);
    gat_bias_act<<<gN128, 256, 0, stream>>>(out, b2, out, N, 0);
}